// InverseLoss_48266842472946
// MI455X (gfx1250) — compile-verified
//
#include <hip/hip_runtime.h>
#include <stdint.h>

// Problem geometry
#define Hh   2048
#define Ww   2048
#define OUTH 2046
#define OUTW 2046
// Tile geometry: 32x64 outputs per block, 34x66 input halo tile
#define TH 32
#define TW 64
#define LH 34
#define LW 66
#define NPT (LH * LW)          // 2244 points per tile
#define NT 256
enum { NITER = (NPT + NT - 1) / NT };   // 9

typedef uint32_t v4u __attribute__((ext_vector_type(4)));
typedef int32_t  v8i __attribute__((ext_vector_type(8)));
typedef int32_t  v4i __attribute__((ext_vector_type(4)));

__device__ __forceinline__ float wave_sum(float v) {
  #pragma unroll
  for (int off = 16; off > 0; off >>= 1) v += __shfl_down(v, off, 32);
  return v;
}

#if __has_builtin(__builtin_amdgcn_tensor_load_to_lds)
// 2D TDM load: tile (tile0 x tile1) elements of 4B data from a row-major
// tensor with row stride `stride0` (elements); OOB (>= td0/td1) zero-fills.
__device__ __forceinline__ void tdm_load_2d(uint32_t lds_off, const void* gptr,
                                            uint32_t td0, uint32_t td1,
                                            uint32_t tile0, uint32_t tile1,
                                            uint32_t stride0) {
  uint64_t ga = (uint64_t)(uintptr_t)gptr;
  v4u g0;
  g0.x = 1u;                                                 // count=1 (valid D#)
  g0.y = lds_off;                                            // lds_addr (bytes)
  g0.z = (uint32_t)ga;                                       // global_addr[31:0]
  g0.w = (uint32_t)((ga >> 32) & 0x01FFFFFFu) | (2u << 30);  // ga[56:32] | type=2
  v8i g1;
  g1[0] = (int)(2u << 16);                                   // data_size = 4B
  g1[1] = (int)((td0 & 0xFFFFu) << 16);                      // tensor_dim0[15:0]
  g1[2] = (int)((td0 >> 16) | ((td1 & 0xFFFFu) << 16));      // dim0 hi | dim1 lo
  g1[3] = (int)((td1 >> 16) | (tile0 << 16));                // dim1 hi | tile_dim0
  g1[4] = (int)tile1;                                        // tile_dim1, tile_dim2=0
  g1[5] = (int)stride0;                                      // tensor_dim0_stride lo
  g1[6] = 0;
  g1[7] = 0;
  v4i g2 = {0, 0, 0, 0};
  v4i g3 = {0, 0, 0, 0};
#if defined(__clang_major__) && (__clang_major__ >= 23)
  v8i g4 = {0, 0, 0, 0, 0, 0, 0, 0};
  __builtin_amdgcn_tensor_load_to_lds(g0, g1, g2, g3, g4, 0);
#else
  __builtin_amdgcn_tensor_load_to_lds(g0, g1, g2, g3, 0);
#endif
}
#endif

__global__ __launch_bounds__(NT)
void stress_stencil_kernel(const float* __restrict__ predE,
                           const float* __restrict__ predV,
                           const float* __restrict__ strain,
                           float* __restrict__ part) {
  __shared__ float sE [NPT];        //  8976 B
  __shared__ float sV [NPT];        //  8976 B
  __shared__ float sS [3 * NPT];    // 26928 B: raw strain tile, then stress planes
  __shared__ float red[8][4];

  const int tid = threadIdx.x;
  const int j0  = blockIdx.x * TW;
  const int i0  = blockIdx.y * TH;

  // ---- Phase 0: TDM DMAs of the full input stream into LDS (wave 0) ----
#if __has_builtin(__builtin_amdgcn_tensor_load_to_lds)
  if ((tid >> 5) == 0) {
    const size_t base = (size_t)i0 * Ww + (size_t)j0;
    tdm_load_2d((uint32_t)(uintptr_t)&sE[0], predE + base,
                (uint32_t)(Ww - j0), (uint32_t)(Hh - i0), LW, LH, Ww);
    tdm_load_2d((uint32_t)(uintptr_t)&sV[0], predV + base,
                (uint32_t)(Ww - j0), (uint32_t)(Hh - i0), LW, LH, Ww);
    tdm_load_2d((uint32_t)(uintptr_t)&sS[0], strain + 3 * base,
                (uint32_t)(3 * (Ww - j0)), (uint32_t)(Hh - i0),
                3 * LW, LH, 3 * Ww);
#if __has_builtin(__builtin_amdgcn_s_wait_tensorcnt)
    __builtin_amdgcn_s_wait_tensorcnt(0);
#endif
  }
#else
  for (int idx = tid; idx < NPT; idx += NT) {
    int r = idx / LW, c = idx - r * LW;
    int gr = i0 + r, gc = j0 + c;
    bool in = (gr < Hh && gc < Ww);
    size_t g = (size_t)gr * Ww + gc;
    sE[idx] = in ? predE[g] : 0.0f;
    sV[idx] = in ? predV[g] : 0.0f;
    sS[3 * idx + 0] = in ? strain[3 * g + 0] : 0.0f;
    sS[3 * idx + 1] = in ? strain[3 * g + 1] : 0.0f;
    sS[3 * idx + 2] = in ? strain[3 * g + 2] : 0.0f;
  }
#endif
  __syncthreads();

  // ---- Phase 1: raw -> stress in registers (reads sS before it is reused) ----
  float rxx[NITER], ryy[NITER], rxy[NITER];
  #pragma unroll
  for (int k = 0; k < NITER; ++k) {
    int idx = tid + k * NT;
    float a = 0.0f, b = 0.0f, c = 0.0f;
    if (idx < NPT) {
      float v_   = sV[idx];
      float E    = sE[idx];                    // zero outside grid (TDM OOB fill)
      float exx  = sS[3 * idx + 0];
      float eyy  = sS[3 * idx + 1];
      float exy  = sS[3 * idx + 2];
      float frac = E / (1.0f - v_ * v_);
      a = (exx + v_ * eyy) * frac;
      b = (v_ * exx + eyy) * frac;
      c = (exy * (1.0f - v_) * 0.5f) * frac;
    }
    rxx[k] = a; ryy[k] = b; rxy[k] = c;
  }
  __syncthreads();                             // WAR: all raw reads done

  // ---- Phase 2: stress planes overwrite the strain buffer in place ----
  float* const sxx = &sS[0];
  float* const syy = &sS[NPT];
  float* const sxy = &sS[2 * NPT];
  #pragma unroll
  for (int k = 0; k < NITER; ++k) {
    int idx = tid + k * NT;
    if (idx < NPT) { sxx[idx] = rxx[k]; syy[idx] = ryy[k]; sxy[idx] = rxy[k]; }
  }
  __syncthreads();

  // ---- Phase 3: separable 3x3 stencil with rolling row sums ----
  //   A = sxx row 3-sum    B = sxy[oj]-sxy[oj+2]
  //   C = syy[oj]-syy[oj+2]  D = sxy row 3-sum   Es = sE row 3-sum
  //   fx = A(r+2)-A(r) + B(r)+B(r+1)+B(r+2)
  //   fy = C(r)+C(r+1)+C(r+2) + D(r+2)-D(r)
  const int oj  = tid & (TW - 1);
  const int oi0 = (tid >> 6) * 8;
  float ax = 0.0f, ay = 0.0f, ae = 0.0f;

  float A0, A1, B0, B1, C0, C1, D0, D1, E0, E1, S0, S1;
  {
    int b = oi0 * LW + oj;
    float p0 = sxy[b], p1 = sxy[b + 1], p2 = sxy[b + 2];
    A0 = sxx[b] + sxx[b + 1] + sxx[b + 2];
    B0 = p0 - p2;
    C0 = syy[b] - syy[b + 2];
    D0 = p0 + p1 + p2;
    S0 = sE[b];
    E0 = S0 + sE[b + 1] + sE[b + 2];
  }
  {
    int b = (oi0 + 1) * LW + oj;
    float p0 = sxy[b], p1 = sxy[b + 1], p2 = sxy[b + 2];
    A1 = sxx[b] + sxx[b + 1] + sxx[b + 2];
    B1 = p0 - p2;
    C1 = syy[b] - syy[b + 2];
    D1 = p0 + p1 + p2;
    S1 = sE[b];
    E1 = S1 + sE[b + 1] + sE[b + 2];
  }

  #pragma unroll
  for (int k = 0; k < 8; ++k) {
    int b2 = (oi0 + k + 2) * LW + oj;
    float q0 = sxy[b2], q1 = sxy[b2 + 1], q2 = sxy[b2 + 2];
    float A2 = sxx[b2] + sxx[b2 + 1] + sxx[b2 + 2];
    float B2 = q0 - q2;
    float C2 = syy[b2] - syy[b2 + 2];
    float D2 = q0 + q1 + q2;
    float e0 = sE[b2];
    float E2 = e0 + sE[b2 + 1] + sE[b2 + 2];

    ae += S0;                                  // sum(pred_E) over owned region
    if ((i0 + oi0 + k) < OUTH && (j0 + oj) < OUTW) {
      float Ec   = E0 + E1 + E2;
      float rinv = 1.0f / Ec;                  // pred_E > 0 -> Ec > 0
      float fx   = A2 - A0 + B0 + B1 + B2;
      float fy   = C0 + C1 + C2 + D2 - D0;
      ax += fabsf(fx * rinv);
      ay += fabsf(fy * rinv);
    }
    A0 = A1; A1 = A2; B0 = B1; B1 = B2; C0 = C1; C1 = C2;
    D0 = D1; D1 = D2; E0 = E1; E1 = E2; S0 = S1; S1 = e0;
  }

  // ---- Phase 4: deterministic block reduction -> per-block partials ----
  ax = wave_sum(ax); ay = wave_sum(ay); ae = wave_sum(ae);
  int lane = tid & 31, w = tid >> 5;
  if (lane == 0) { red[w][0] = ax; red[w][1] = ay; red[w][2] = ae; }
  __syncthreads();
  if (tid == 0) {
    float sx = 0.0f, sy = 0.0f, se = 0.0f;
    #pragma unroll
    for (int i = 0; i < 8; ++i) { sx += red[i][0]; sy += red[i][1]; se += red[i][2]; }
    int bid = blockIdx.y * gridDim.x + blockIdx.x;
    part[3 * bid + 0] = sx;
    part[3 * bid + 1] = sy;
    part[3 * bid + 2] = se;
  }
}

__global__ __launch_bounds__(256)
void finalize_kernel(const float* __restrict__ part, float* __restrict__ out, int nb) {
  __shared__ float red[8][4];
  float sx = 0.0f, sy = 0.0f, se = 0.0f;
  for (int i = threadIdx.x; i < nb; i += 256) {
    sx += part[3 * i + 0];
    sy += part[3 * i + 1];
    se += part[3 * i + 2];
  }
  sx = wave_sum(sx); sy = wave_sum(sy); se = wave_sum(se);
  int lane = threadIdx.x & 31, w = threadIdx.x >> 5;
  if (lane == 0) { red[w][0] = sx; red[w][1] = sy; red[w][2] = se; }
  __syncthreads();
  if (threadIdx.x == 0) {
    float tx = 0.0f, ty = 0.0f, te = 0.0f;
    #pragma unroll
    for (int i = 0; i < 8; ++i) { tx += red[i][0]; ty += red[i][1]; te += red[i][2]; }
    const float invn = 1.0f / ((float)OUTW * (float)OUTH);
    float loss = tx * invn + ty * invn +
                 fabsf(te / ((float)Hh * (float)Ww) - 1.0f) * 0.01f;
    out[0] = loss;
  }
}

extern "C" void kernel_launch(void* const* d_in, const int* in_sizes, int n_in,
                              void* d_out, int out_size, void* d_ws, size_t ws_size,
                              hipStream_t stream) {
  const float* predE  = (const float*)d_in[0];
  const float* predV  = (const float*)d_in[1];
  const float* strain = (const float*)d_in[2];
  float* part = (float*)d_ws;                 // 2048 blocks * 3 floats = 24 KB
  dim3 grid(Ww / TW, Hh / TH);                // 32 x 64 = 2048 blocks
  stress_stencil_kernel<<<grid, NT, 0, stream>>>(predE, predV, strain, part);
  finalize_kernel<<<1, 256, 0, stream>>>(part, (float*)d_out, (int)(grid.x * grid.y));
}